// RGCN_240518168950
// MI455X (gfx1250) — compile-verified
//
#include <hip/hip_runtime.h>
#include <hip/hip_bf16.h>
#include <math.h>

// ---------------------------------------------------------------------------
// RGCN (3-layer) for MI455X / gfx1250.
//  - Edges bucketed by relation (counting sort, re-done every launch).
//  - 16-edge tiles: gather x[src] rows -> V_WMMA_F32_16X16X4_F32 against
//    Wrel[r] -> global_atomic_add_f32 scatter into the 51MB (L2-resident)
//    accumulator. Root term is a dense WMMA GEMM fused with the bias.
//  - DOUT templated so all weight/output addressing strength-reduces to
//    immediate offsets on global_load/global_store.
// ---------------------------------------------------------------------------

typedef float v2f __attribute__((ext_vector_type(2)));
typedef float v8f __attribute__((ext_vector_type(8)));

#define WMMA_F32(a, b, c) \
  __builtin_amdgcn_wmma_f32_16x16x4_f32(false, (a), false, (b), (short)0, (c), false, false)

static constexpr int D_IN = 128;   // feature width of x (and stride of node buffers)

// ------------------------------ bucketing ----------------------------------

__global__ void k_zero_ints(int* __restrict__ p, int n) {
  int i = blockIdx.x * blockDim.x + threadIdx.x;
  if (i < n) p[i] = 0;
}

__global__ void k_hist(const int* __restrict__ etype, int E, int* __restrict__ counts) {
  int e = blockIdx.x * blockDim.x + threadIdx.x;
  if (e < E) atomicAdd(&counts[etype[e] & 7], 1);
}

__global__ void k_scan(const int* __restrict__ counts, int* __restrict__ starts) {
  if (threadIdx.x == 0 && blockIdx.x == 0) {
    int acc = 0;
    for (int r = 0; r < 8; ++r) {
      starts[r] = acc;
      acc += (counts[r] + 15) & ~15;   // pad each bucket to a 16-edge tile
    }
    starts[8] = acc;
  }
}

__global__ void k_scatter(const int* __restrict__ ei, const int* __restrict__ etype, int E,
                          const int* __restrict__ starts, int* __restrict__ cursor,
                          int* __restrict__ bsrc, int* __restrict__ bdst) {
  int e = blockIdx.x * blockDim.x + threadIdx.x;
  if (e >= E) return;
  int r = etype[e] & 7;
  int pos = atomicAdd(&cursor[r], 1);
  int idx = starts[r] + pos;
  bsrc[idx] = ei[e];        // src row
  bdst[idx] = ei[E + e];    // dst row
}

__global__ void k_pad(const int* __restrict__ counts, const int* __restrict__ starts,
                      int* __restrict__ bsrc, int* __restrict__ bdst, int dummyRow) {
  int r = blockIdx.x;
  int lo = starts[r] + counts[r];
  int hi = starts[r + 1];
  for (int i = lo + (int)threadIdx.x; i < hi; i += blockDim.x) {
    bsrc[i] = 0;
    bdst[i] = dummyRow;     // padding lands in a scratch row, never read back
  }
}

// --------------------------- dense root GEMM -------------------------------
// y[row, col] = bias[col] + sum_k x[row,k] * W[k,col]
// grid.x = node tiles (16 rows each); blockDim = 32 * (DOUT/16); 1 wave = 1 tile.

template <int DOUT>
__global__ __launch_bounds__(32 * (DOUT / 16)) void k_root_gemm(
    const float* __restrict__ x, const float* __restrict__ W,
    const float* __restrict__ bias, float* __restrict__ y, int N) {
  const int wave = threadIdx.x >> 5, lane = threadIdx.x & 31;
  const int m = lane & 15;
  const int khalf = (lane >> 4) << 1;              // 0 for lanes 0-15, 2 for 16-31
  const int row = (blockIdx.x << 4) + m;
  const int rowc = row < N ? row : N - 1;          // clamp A loads (stores guarded)
  const float* xa = x + (size_t)rowc * D_IN;
  const int col = (wave << 4) + m;
  const float* wb = W + col;
  v8f c = {};
#pragma unroll 8
  for (int k = 0; k < D_IN; k += 4) {
    v2f a, b;
    a.x = xa[k + khalf];
    a.y = xa[k + khalf + 1];
    b.x = wb[(k + khalf) * DOUT];                  // immediate-offset loads
    b.y = wb[(k + khalf + 1) * DOUT];
    c = WMMA_F32(a, b, c);
  }
  const float bv = bias[col];
  const int rbase = (blockIdx.x << 4) + ((lane >> 4) << 3);   // M = rbase..rbase+7
  float* yp = y + (size_t)rbase * DOUT + col;
  if ((blockIdx.x << 4) + 16 <= N) {
    // uniform full-tile fast path: 8 unguarded stores, immediate offsets
#pragma unroll
    for (int i = 0; i < 8; ++i) yp[i * DOUT] = c[i] + bv;
  } else {
#pragma unroll
    for (int i = 0; i < 8; ++i)
      if (rbase + i < N) yp[i * DOUT] = c[i] + bv;
  }
}

// --------------------------- edge message WMMA -----------------------------
// One 16-edge tile per block; wave w handles output columns [16w, 16w+16).
// C(16 x DOUT) = x[src rows](16 x 128) @ Wrel[r](128 x DOUT); atomic scatter.

template <int DOUT>
__global__ __launch_bounds__(32 * (DOUT / 16)) void k_edge_wmma(
    const float* __restrict__ x, const float* __restrict__ Wrel,
    const int* __restrict__ bsrc, const int* __restrict__ bdst,
    const int* __restrict__ starts, float* __restrict__ y) {
  const int base = blockIdx.x << 4;
  if (base >= starts[8]) return;                   // uniform early-exit (EXEC stays full)
  const int wave = threadIdx.x >> 5, lane = threadIdx.x & 31;
  int r = 0;
#pragma unroll
  for (int i = 1; i < 8; ++i) r += (base >= starts[i]) ? 1 : 0;

  const int m = lane & 15;
  const int khalf = (lane >> 4) << 1;
  const int srow = bsrc[base + m];                 // gathered source row (L2-resident x)
  const float* xa = x + (size_t)srow * D_IN;
  const int col = (wave << 4) + m;
  const float* wb = Wrel + (size_t)r * (D_IN * DOUT) + col;

  v8f c = {};
#pragma unroll 8
  for (int k = 0; k < D_IN; k += 4) {
    v2f a, b;
    a.x = xa[k + khalf];
    a.y = xa[k + khalf + 1];
    b.x = wb[(k + khalf) * DOUT];
    b.y = wb[(k + khalf + 1) * DOUT];
    c = WMMA_F32(a, b, c);
  }

  const int mBase = (lane >> 4) << 3;              // rows 0-7 or 8-15 of the tile
#pragma unroll
  for (int i = 0; i < 8; ++i) {
    int drow = bdst[base + mBase + i];
    atomicAdd(&y[(size_t)drow * DOUT + col], c[i]);
  }
}

// ------------------------------ pointwise ----------------------------------

__global__ void k_relu4(float4* __restrict__ p, long long n4) {
  long long i = (long long)blockIdx.x * blockDim.x + threadIdx.x;
  if (i < n4) {
    float4 v = p[i];
    v.x = fmaxf(v.x, 0.0f);
    v.y = fmaxf(v.y, 0.0f);
    v.z = fmaxf(v.z, 0.0f);
    v.w = fmaxf(v.w, 0.0f);
    p[i] = v;
  }
}

// dOut == 64: one wave per node, 2 values per lane, butterfly reductions.
__global__ void k_logsoftmax(const float* __restrict__ y, float* __restrict__ out, int N) {
  int wave = threadIdx.x >> 5, lane = threadIdx.x & 31;
  int node = blockIdx.x * (blockDim.x >> 5) + wave;
  if (node >= N) return;
  const float* row = y + (size_t)node * 64;
  float v0 = row[lane], v1 = row[lane + 32];
  float mx = fmaxf(v0, v1);
#pragma unroll
  for (int off = 16; off > 0; off >>= 1) mx = fmaxf(mx, __shfl_xor(mx, off, 32));
  float s = expf(v0 - mx) + expf(v1 - mx);
#pragma unroll
  for (int off = 16; off > 0; off >>= 1) s += __shfl_xor(s, off, 32);
  float lse = logf(s) + mx;
  float* o = out + (size_t)node * 64;
  o[lane] = v0 - lse;
  o[lane + 32] = v1 - lse;
}

// ------------------------------ launcher -----------------------------------

static inline void launch_layer(const float* x, const float* Wrel, const float* Wroot,
                                const float* bias, float* y, int dOut, int N,
                                const int* bsrc, const int* bdst, const int* starts,
                                int nodeTiles, int edgeTilesMax, hipStream_t stream) {
  if (dOut == 128) {
    k_root_gemm<128><<<nodeTiles, 32 * (128 / 16), 0, stream>>>(x, Wroot, bias, y, N);
    k_edge_wmma<128><<<edgeTilesMax, 32 * (128 / 16), 0, stream>>>(x, Wrel, bsrc, bdst, starts, y);
  } else {
    k_root_gemm<64><<<nodeTiles, 32 * (64 / 16), 0, stream>>>(x, Wroot, bias, y, N);
    k_edge_wmma<64><<<edgeTilesMax, 32 * (64 / 16), 0, stream>>>(x, Wrel, bsrc, bdst, starts, y);
  }
}

extern "C" void kernel_launch(void* const* d_in, const int* in_sizes, int n_in,
                              void* d_out, int out_size, void* d_ws, size_t ws_size,
                              hipStream_t stream) {
  const float* x0     = (const float*)d_in[0];
  const int*   ei     = (const int*)  d_in[1];   // [2, E] flat: src then dst
  const int*   etype  = (const int*)  d_in[2];
  const float* Wrel1  = (const float*)d_in[3];
  const float* Wroot1 = (const float*)d_in[4];
  const float* b1     = (const float*)d_in[5];
  const float* Wrel2  = (const float*)d_in[6];
  const float* Wroot2 = (const float*)d_in[7];
  const float* b2     = (const float*)d_in[8];
  const float* Wrel3  = (const float*)d_in[9];
  const float* Wroot3 = (const float*)d_in[10];
  const float* b3     = (const float*)d_in[11];

  const int N  = in_sizes[0] / D_IN;   // 100000
  const int E  = in_sizes[2];          // 1600000
  const int D1 = 128, D2 = 128;
  const int D3 = in_sizes[11];         // 64

  // workspace carve-up
  char* w = (char*)d_ws;
  int* counts = (int*)w;               // [8]
  int* cursor = counts + 8;            // [8]
  int* starts = cursor + 8;            // [9]
  size_t off = 256;
  int* bsrc = (int*)(w + off); off += (size_t)(E + 256) * sizeof(int);
  int* bdst = (int*)(w + off); off += (size_t)(E + 256) * sizeof(int);
  off = (off + 255) & ~(size_t)255;
  float* bufA = (float*)(w + off); off += (size_t)(N + 16) * D_IN * sizeof(float);
  float* bufB = (float*)(w + off);

  // ---- bucket edges by relation (fresh every call; deterministic work) ----
  k_zero_ints<<<1, 64, 0, stream>>>(counts, 32);
  k_hist    <<<(E + 255) / 256, 256, 0, stream>>>(etype, E, counts);
  k_scan    <<<1, 1, 0, stream>>>(counts, starts);
  k_scatter <<<(E + 255) / 256, 256, 0, stream>>>(ei, etype, E, starts, cursor, bsrc, bdst);
  k_pad     <<<8, 64, 0, stream>>>(counts, starts, bsrc, bdst, N);

  const int nodeTiles    = (N + 15) / 16;
  const int edgeTilesMax = E / 16 + 8;      // upper bound on padded tiles

  // ---- layer 1: x0 -> bufA, relu in place ----
  launch_layer(x0, Wrel1, Wroot1, b1, bufA, D1, N, bsrc, bdst, starts,
               nodeTiles, edgeTilesMax, stream);
  {
    long long n4 = (long long)N * D1 / 4;
    k_relu4<<<(unsigned)((n4 + 255) / 256), 256, 0, stream>>>((float4*)bufA, n4);
  }

  // ---- layer 2: bufA -> bufB, relu in place ----
  launch_layer(bufA, Wrel2, Wroot2, b2, bufB, D2, N, bsrc, bdst, starts,
               nodeTiles, edgeTilesMax, stream);
  {
    long long n4 = (long long)N * D2 / 4;
    k_relu4<<<(unsigned)((n4 + 255) / 256), 256, 0, stream>>>((float4*)bufB, n4);
  }

  // ---- layer 3: bufB -> bufA (stride D3), then log_softmax -> d_out ----
  launch_layer(bufB, Wrel3, Wroot3, b3, bufA, D3, N, bsrc, bdst, starts,
               nodeTiles, edgeTilesMax, stream);
  k_logsoftmax<<<(N + 7) / 8, 256, 0, stream>>>(bufA, (float*)d_out, N);
}